// De_conv_batched_for_scan_43104291782951
// MI455X (gfx1250) — compile-verified
//
#include <hip/hip_runtime.h>

// Scalar supervoxel-edge loss:  out = loss_old + (1/(B*N)) * sum_patches sum_{6x6 interior} ex*ey*edge
//   ex = f(m[ix,iy]*m[ix+1,iy]),  ey = f(m[ix,iy]*m[ix,iy+1]),  f(a) = 2a - a^2
// Memory-bound: ~272 MB effective HBM traffic -> ~12us floor on MI455X (23.3 TB/s).

typedef __attribute__((ext_vector_type(2))) float v2f;
typedef __attribute__((ext_vector_type(8))) float v8f;

#if defined(__gfx1250__) && __has_builtin(__builtin_amdgcn_wmma_f32_16x16x4_f32)
#define SV_USE_WMMA_REDUCE 1
#endif

namespace {
constexpr int kX = 1024, kY = 1024, kB = 16;
constexpr int kNXP = 129;                    // (1024+8)/8 patches per axis (all shift configs)
constexpr int kNPB = kNXP * kNXP;            // 16641 patches per batch image
constexpr int kTot = kB * kNPB;              // 266256 total patches == mean divisor
constexpr int kThreads = 256;
constexpr int kBlocks = (kTot + kThreads - 1) / kThreads;  // 1041
}  // namespace

// Per-patch loss. mb = &mask[b,0,0,mi] (stride 4 floats in y), eb = &edge[b,0,0,0].
// GUARD=false: patch fully inside the original image (hot path, no bounds checks,
// vectorized edge loads). GUARD=true: padded-border patches, per-element clamping.
template <bool GUARD>
__device__ __forceinline__ float patch_loss(const float* __restrict__ mb,
                                            const float* __restrict__ eb,
                                            int x0, int y0) {
  float cur[7];
  auto load_row = [&](int ix, float* r) {
    const int x = x0 + ix;
    if constexpr (GUARD) {
      const bool rv = (unsigned)x < (unsigned)kX;
#pragma unroll
      for (int iy = 0; iy < 7; ++iy) {
        const int y = y0 + iy;
        const bool v = rv && ((unsigned)y < (unsigned)kY);
        r[iy] = v ? mb[(x * kY + y) * 4] : 0.0f;
      }
    } else {
#pragma unroll
      for (int iy = 0; iy < 7; ++iy) r[iy] = mb[(x * kY + y0 + iy) * 4];
    }
  };

  load_row(0, cur);
  float acc = 0.0f;
#pragma unroll
  for (int ix = 0; ix < 6; ++ix) {
    float nxt[7];
    load_row(ix + 1, nxt);
    const int x = x0 + ix;
    float e[6];
    if constexpr (GUARD) {
      const bool rv = (unsigned)x < (unsigned)kX;
#pragma unroll
      for (int iy = 0; iy < 6; ++iy) {
        const int y = y0 + iy;
        e[iy] = (rv && ((unsigned)y < (unsigned)kY)) ? eb[x * kY + y] : 0.0f;
      }
    } else {
      // y0 is a multiple of 4 floats -> 16B aligned; edge channel dim is 1 (contiguous).
      const float4 e4 = *reinterpret_cast<const float4*>(eb + x * kY + y0);
      const float2 e2 = *reinterpret_cast<const float2*>(eb + x * kY + y0 + 4);
      e[0] = e4.x; e[1] = e4.y; e[2] = e4.z; e[3] = e4.w; e[4] = e2.x; e[5] = e2.y;
    }
#pragma unroll
    for (int iy = 0; iy < 6; ++iy) {
      const float c = cur[iy] * cur[iy + 1];   // along patch-y
      const float ey = c * (2.0f - c);         // 2c - c^2
      const float a = cur[iy] * nxt[iy];       // along patch-x
      const float ex = a * (2.0f - a);
      acc = fmaf(ex * ey, e[iy], acc);
    }
#pragma unroll
    for (int iy = 0; iy < 7; ++iy) cur[iy] = nxt[iy];
  }
  return acc;
}

__global__ __launch_bounds__(kThreads) void sv_loss_partial_kernel(
    const float* __restrict__ mask, const float* __restrict__ edge,
    const int* __restrict__ mask_index_p, float* __restrict__ partials) {
  const int tid = blockIdx.x * kThreads + threadIdx.x;
  const int mi = *mask_index_p;          // uniform -> scalar load
  const int sx = mi & 1;                 // SHIFTS = [(0,0),(1,0),(0,1),(1,1)]
  const int sy = (mi >> 1) & 1;

  float acc = 0.0f;
  if (tid < kTot) {
    const int b = tid / kNPB;
    const int n = tid - b * kNPB;
    const int pxi = n / kNXP;
    const int pyi = n - pxi * kNXP;
    const int x0 = pxi * 8 - sx * 4;     // original-image coords of patch row 0
    const int y0 = pyi * 8 - sy * 4;
    const float* mb = mask + (size_t)b * ((size_t)kX * kY * 4) + mi;
    const float* eb = edge + (size_t)b * ((size_t)kX * kY);
    const bool interior =
        (x0 >= 0) && (x0 + 6 < kX) && (y0 >= 0) && (y0 + 6 < kY);
    acc = interior ? patch_loss<false>(mb, eb, x0, y0)
                   : patch_loss<true>(mb, eb, x0, y0);
  }

  // Deterministic block reduction (fixed tree order).
  __shared__ float red[kThreads];
  red[threadIdx.x] = acc;
  __syncthreads();
#pragma unroll
  for (int s = kThreads / 2; s > 0; s >>= 1) {
    if (threadIdx.x < s) red[threadIdx.x] += red[threadIdx.x + s];
    __syncthreads();
  }
  if (threadIdx.x == 0) partials[blockIdx.x] = red[0];
}

// Single-wave (32 lanes, EXEC all-ones) finalize: serial grid-stride partial sums,
// then the 32 lane values are reduced with one V_WMMA_F32_16X16X4_F32 against an
// all-ones B matrix (D rows = row-sums of A), falling back to shuffles if the
// builtin is unavailable.
__global__ __launch_bounds__(32) void sv_loss_finalize_kernel(
    const float* __restrict__ partials, int n,
    const float* __restrict__ loss_old, float* __restrict__ out) {
  const int l = threadIdx.x;
  float s = 0.0f;
  for (int i = l; i < n; i += 32) s += partials[i];
  constexpr float inv = (float)(1.0 / (double)kTot);

#ifdef SV_USE_WMMA_REDUCE
  __shared__ float red[32];
  red[l] = s;
  __syncthreads();
  // A (16x4 f32): lanes 0-15 hold K=0,1 in v[0],v[1]; lanes 16-31 hold K=2,3.
  // Fill A[i][0]=red[2i], A[i][1]=red[2i+1], A[i][2]=A[i][3]=0.
  v2f a;
  if (l < 16) { a.x = red[2 * l]; a.y = red[2 * l + 1]; }
  else        { a.x = 0.0f;       a.y = 0.0f; }
  v2f ones; ones.x = 1.0f; ones.y = 1.0f;      // all-ones B: layout-agnostic
  v8f c = {0.0f, 0.0f, 0.0f, 0.0f, 0.0f, 0.0f, 0.0f, 0.0f};
  v8f d = __builtin_amdgcn_wmma_f32_16x16x4_f32(
      /*neg_a=*/false, a, /*neg_b=*/false, ones,
      /*c_mod=*/(short)0, c, /*reuse_a=*/false, /*reuse_b=*/false);
  // D[i][j] = rowsum_i for every column j. Lane 0 holds M=0..7 in v0..v7,
  // lane 16 holds M=8..15 -> total = lanesum(0) + lanesum(16).
  float ls = d[0] + d[1] + d[2] + d[3] + d[4] + d[5] + d[6] + d[7];
  const float total = __shfl(ls, 0, 32) + __shfl(ls, 16, 32);
#else
  float v = s;
  for (int off = 16; off > 0; off >>= 1) v += __shfl_down(v, off, 32);
  const float total = __shfl(v, 0, 32);
#endif

  if (l == 0) out[0] = loss_old[0] + total * inv;
}

extern "C" void kernel_launch(void* const* d_in, const int* in_sizes, int n_in,
                              void* d_out, int out_size, void* d_ws, size_t ws_size,
                              hipStream_t stream) {
  (void)in_sizes; (void)n_in; (void)out_size; (void)ws_size;
  // setup_inputs order: resized_image (dead code), mask_combined, edge_map,
  //                     loss_old, mask_index
  const float* mask      = (const float*)d_in[1];
  const float* edge      = (const float*)d_in[2];
  const float* loss_old  = (const float*)d_in[3];
  const int*   mask_idx  = (const int*)d_in[4];
  float* out      = (float*)d_out;
  float* partials = (float*)d_ws;  // kBlocks floats (~4.1 KB)

  sv_loss_partial_kernel<<<dim3(kBlocks), dim3(kThreads), 0, stream>>>(
      mask, edge, mask_idx, partials);
  sv_loss_finalize_kernel<<<dim3(1), dim3(32), 0, stream>>>(
      partials, kBlocks, loss_old, out);
}